// CellLineMLPPredictor_16295105921293
// MI455X (gfx1250) — compile-verified
//
#include <hip/hip_runtime.h>
#include <hip/hip_bf16.h>

typedef float v2f __attribute__((ext_vector_type(2)));
typedef float v8f __attribute__((ext_vector_type(8)));

// ---------------------------------------------------------------------------
// Kernel 1: build X[512][2048] = concat(h_drug[p0], attrs[:,1], h_drug[p1], attrs[:,3])
// ---------------------------------------------------------------------------
__global__ __launch_bounds__(256)
void build_input_kernel(const int* __restrict__ pairs,
                        const float* __restrict__ attrs,
                        const float* __restrict__ h_drug,
                        float* __restrict__ X)
{
    const int b = blockIdx.x;
    const int t = threadIdx.x;
    const int p0 = pairs[2 * b + 0];
    const int p1 = pairs[2 * b + 1];
    const float* r0 = h_drug + (size_t)p0 * 1023;
    const float* r1 = h_drug + (size_t)p1 * 1023;
    float* xb = X + (size_t)b * 2048;
    for (int i = t; i < 1023; i += 256) {
        xb[i]        = r0[i];
        xb[1024 + i] = r1[i];
    }
    if (t == 0) {
        xb[1023] = attrs[4 * b + 1];   // attrs[:, n_attr-1 : n_attr] with n_attr=2
        xb[2047] = attrs[4 * b + 3];   // attrs[:, -1:]
    }
}

// ---------------------------------------------------------------------------
// Kernel 2: Y[M,N] = act(X[M,K] @ W[N,K]^T + bias[N]) via V_WMMA_F32_16X16X4_F32.
// Block = 256 threads (8 waves). Block tile 64(M) x 128(N).
// Wave w: m-strip = (w&3)*16, n-strip = (w>>2)*64 -> 4 x (16x16) accumulators.
// f32 A 16x4 layout: lane<16 holds (M=lane, K=k..k+1), lane>=16 (M=lane-16, K=k+2..k+3).
// C/D layout: n = lane%16, m = r + 8*(lane/16) for VGPR r of 8.
// ---------------------------------------------------------------------------
__global__ __launch_bounds__(256)
void gemm_bias_relu_kernel(const float* __restrict__ X,
                           const float* __restrict__ W,
                           const float* __restrict__ bias,
                           float* __restrict__ Y,
                           int M, int N, int K, int do_relu)
{
    const int tid  = threadIdx.x;
    const int w    = tid >> 5;
    const int lane = tid & 31;
    const int llo  = lane & 15;
    const int lhi  = lane >> 4;

    const int m0 = blockIdx.y * 64  + (w & 3) * 16;
    const int n0 = blockIdx.x * 128 + (w >> 2) * 64;

    const float* arow  = X + (size_t)(m0 + llo) * K;
    const float* brow0 = W + (size_t)(n0 +  0 + llo) * K;
    const float* brow1 = W + (size_t)(n0 + 16 + llo) * K;
    const float* brow2 = W + (size_t)(n0 + 32 + llo) * K;
    const float* brow3 = W + (size_t)(n0 + 48 + llo) * K;

    // D = A*B + C lets us seed the accumulator with the bias (constant along m).
    const float bb0 = bias[n0 +  0 + llo];
    const float bb1 = bias[n0 + 16 + llo];
    const float bb2 = bias[n0 + 32 + llo];
    const float bb3 = bias[n0 + 48 + llo];
    v8f acc0, acc1, acc2, acc3;
#pragma unroll
    for (int r = 0; r < 8; ++r) { acc0[r] = bb0; acc1[r] = bb1; acc2[r] = bb2; acc3[r] = bb3; }

    for (int k = 0; k < K; k += 4) {
        if ((k & 63) == 0) {   // stream ~1KB ahead (global_prefetch_b8); L2 holds everything
            __builtin_prefetch(arow  + k + 256, 0, 1);
            __builtin_prefetch(brow0 + k + 256, 0, 1);
            __builtin_prefetch(brow1 + k + 256, 0, 1);
            __builtin_prefetch(brow2 + k + 256, 0, 1);
            __builtin_prefetch(brow3 + k + 256, 0, 1);
        }
        v2f a = *(const v2f*)(arow + k + 2 * lhi);
        v2f b0, b1, b2, b3;
        b0.x = brow0[k + lhi]; b0.y = brow0[k + 2 + lhi];
        b1.x = brow1[k + lhi]; b1.y = brow1[k + 2 + lhi];
        b2.x = brow2[k + lhi]; b2.y = brow2[k + 2 + lhi];
        b3.x = brow3[k + lhi]; b3.y = brow3[k + 2 + lhi];
        acc0 = __builtin_amdgcn_wmma_f32_16x16x4_f32(false, a, false, b0, (short)0, acc0, false, false);
        acc1 = __builtin_amdgcn_wmma_f32_16x16x4_f32(false, a, false, b1, (short)0, acc1, false, false);
        acc2 = __builtin_amdgcn_wmma_f32_16x16x4_f32(false, a, false, b2, (short)0, acc2, false, false);
        acc3 = __builtin_amdgcn_wmma_f32_16x16x4_f32(false, a, false, b3, (short)0, acc3, false, false);
    }

    float* yout = Y + (size_t)(m0 + 8 * lhi) * N + (n0 + llo);
#pragma unroll
    for (int r = 0; r < 8; ++r) {
        float v0 = acc0[r], v1 = acc1[r], v2 = acc2[r], v3 = acc3[r];
        if (do_relu) {
            v0 = v0 > 0.f ? v0 : 0.f; v1 = v1 > 0.f ? v1 : 0.f;
            v2 = v2 > 0.f ? v2 : 0.f; v3 = v3 > 0.f ? v3 : 0.f;
        }
        yout[(size_t)r * N +  0] = v0;
        yout[(size_t)r * N + 16] = v1;
        yout[(size_t)r * N + 32] = v2;
        yout[(size_t)r * N + 48] = v3;
    }
}

// ---------------------------------------------------------------------------
// Kernel 3: per-cell-line head. One block per cell line (32 blocks, 512 thr = 16 waves).
// Compacts this cell's sample list, then processes 16 samples at a time as the
// WMMA N dimension: z[512 x 16] = relu(L0[c] (512x1024) @ Y2_chunk^T + O0[c]),
// streaming L0[c] once per chunk. Final scalar: out[s] = L1[c]·z[:,s] + O1[c].
// ---------------------------------------------------------------------------
__global__ __launch_bounds__(512)
void cell_head_kernel(const int* __restrict__ cells,
                      const float* __restrict__ Y2,    // [512,1024]
                      const float* __restrict__ L0,    // [32,512,1024]
                      const float* __restrict__ O0,    // [32,512]
                      const float* __restrict__ L1,    // [32,512]
                      const float* __restrict__ O1,    // [32]
                      float* __restrict__ out,         // [512]
                      int Bn)
{
    __shared__ int   slist[512];
    __shared__ int   scount;
    __shared__ float zbuf[512][17];   // [o][sample], padded stride vs 64 banks

    const int c   = blockIdx.x;
    const int tid = threadIdx.x;

    if (tid == 0) scount = 0;
    __syncthreads();
    if (tid < Bn && cells[tid] == c) {
        int p = atomicAdd(&scount, 1);
        slist[p] = tid;
    }
    __syncthreads();
    const int count = scount;
    if (count == 0) return;                  // block-uniform

    const int w    = tid >> 5;
    const int lane = tid & 31;
    const int llo  = lane & 15;
    const int lhi  = lane >> 4;

    const float* L0c = L0 + (size_t)c * 512 * 1024;
    const float* O0c = O0 + (size_t)c * 512;
    const float* L1c = L1 + (size_t)c * 512;

    for (int base = 0; base < count; base += 16) {
        // Per-lane sample index = WMMA column n; pad slots alias sample slist[0]
        // (valid memory, result discarded).
        const int sslot = base + llo;
        const int sidx  = slist[sslot < count ? sslot : 0];
        const float* yrow = Y2 + (size_t)sidx * 1024;

#pragma unroll
        for (int j = 0; j < 2; ++j) {        // 16 waves x 2 = 32 o-tiles of 16
            const int o0 = (2 * w + j) * 16;
            const float* arow = L0c + (size_t)(o0 + llo) * 1024;
            v8f acc;
#pragma unroll
            for (int r = 0; r < 8; ++r) acc[r] = O0c[o0 + r + 8 * lhi];   // bias along m
            for (int k = 0; k < 1024; k += 4) {
                if ((k & 63) == 0) __builtin_prefetch(arow + k + 256, 0, 1);
                v2f a = *(const v2f*)(arow + k + 2 * lhi);
                v2f b; b.x = yrow[k + lhi]; b.y = yrow[k + 2 + lhi];
                acc = __builtin_amdgcn_wmma_f32_16x16x4_f32(false, a, false, b, (short)0, acc, false, false);
            }
#pragma unroll
            for (int r = 0; r < 8; ++r) {
                float v = acc[r];
                zbuf[o0 + r + 8 * lhi][llo] = v > 0.f ? v : 0.f;
            }
        }
        __syncthreads();

        if (tid < 16 && base + tid < count) {
            float s = O1[c];
            for (int o = 0; o < 512; ++o) s += L1c[o] * zbuf[o][tid];
            out[slist[base + tid]] = s;
        }
        __syncthreads();
    }
}

// ---------------------------------------------------------------------------
extern "C" void kernel_launch(void* const* d_in, const int* in_sizes, int n_in,
                              void* d_out, int out_size, void* d_ws, size_t ws_size,
                              hipStream_t stream)
{
    const int*   pairs = (const int*)d_in[0];
    const int*   cells = (const int*)d_in[1];
    const float* attrs = (const float*)d_in[2];
    const float* hdrug = (const float*)d_in[3];
    const float* W0    = (const float*)d_in[4];
    const float* b0    = (const float*)d_in[5];
    const float* W1    = (const float*)d_in[6];
    const float* b1    = (const float*)d_in[7];
    const float* L0    = (const float*)d_in[8];
    const float* O0    = (const float*)d_in[9];
    const float* L1    = (const float*)d_in[10];
    const float* O1    = (const float*)d_in[11];
    float* out = (float*)d_out;

    const int B = 512, D0 = 2048, D1 = 2048, D2 = 1024;

    float* X  = (float*)d_ws;                 //  4 MB
    float* Y1 = X  + (size_t)B * D0;          //  4 MB
    float* Y2 = Y1 + (size_t)B * D1;          //  2 MB

    build_input_kernel<<<B, 256, 0, stream>>>(pairs, attrs, hdrug, X);
    gemm_bias_relu_kernel<<<dim3(D1 / 128, B / 64), 256, 0, stream>>>(X,  W0, b0, Y1, B, D1, D0, 1);
    gemm_bias_relu_kernel<<<dim3(D2 / 128, B / 64), 256, 0, stream>>>(Y1, W1, b1, Y2, B, D2, D1, 1);
    cell_head_kernel<<<32, 512, 0, stream>>>(cells, Y2, L0, O0, L1, O1, out, B);
}